// Quantize3_43645457662412
// MI455X (gfx1250) — compile-verified
//
#include <hip/hip_runtime.h>

typedef __attribute__((ext_vector_type(2))) float v2f;
typedef __attribute__((ext_vector_type(8))) float v8f;

#define DDIM   256
#define NE     8192
#define NROWS  32768
#define TOPK   64

// d_out layout (flat float elements, reference return order)
#define Q_OFF    0u
#define DIFF_OFF 8388608u
#define IND_OFF  8388609u
#define DIST_OFF 8421377u

#define KCHUNK   32               // K-depth staged per LDS buffer
#define NCHUNK   (DDIM / KCHUNK)  // 8
#define NBLK     128              // N columns per block (8 x 16-wide frags/wave)
#define PRSTRIDE 288              // floats per K-pair row: 2*NBLK + 32-bank skew

// ---------------- row norms: ||x_i||^2, one block per row ----------------
__global__ void row_norm_kernel(const float* __restrict__ x, float* __restrict__ xn) {
  __shared__ float red[256];
  const int row = blockIdx.x;
  float v = x[(size_t)row * DDIM + threadIdx.x];
  red[threadIdx.x] = v * v;
  __syncthreads();
  for (int s = 128; s > 0; s >>= 1) {
    if ((int)threadIdx.x < s) red[threadIdx.x] += red[threadIdx.x + s];
    __syncthreads();
  }
  if (threadIdx.x == 0) xn[row] = red[0];
}

// ---------------- col norms: ||e_j||^2, coalesced across columns ----------------
__global__ void col_norm_kernel(const float* __restrict__ e, float* __restrict__ en) {
  const int col = blockIdx.x * blockDim.x + threadIdx.x;
  float s = 0.0f;
  for (int d = 0; d < DDIM; ++d) {
    float v = e[(size_t)d * NE + col];
    s += v * v;
  }
  en[col] = s;
}

// ---------------- dist GEMM: V_WMMA_F32_16X16X4_F32 + async-LDS double buffer ----
// Block = 256 threads = 8 waves covering 128(M) x 128(N). B tile staged once per
// block via global_load_async_to_lds_b32 (ASYNCcnt), K-pair-INTERLEAVED in LDS:
// element (k,col) at (k>>1)*PRSTRIDE + 2*col + (k&1), so each wave B-fragment
// {B[k][col], B[k+1][col]} is one aligned ds_load_b64 straight into a VGPR pair.
// Double buffering at two levels: LDS chunks (DMA of c+1 overlaps WMMA on c) and
// registers (B frags of k-step s+1 load while the 8 WMMAs of step s execute; all
// A frags for a chunk preload at chunk start).
// A fragment (16x4 f32 ISA layout): lanes 0-15 hold K={k,k+1}, lanes 16-31 K={k+2,k+3}.
__global__ void dist_wmma_kernel(const float* __restrict__ x,
                                 const float* __restrict__ e,
                                 const float* __restrict__ xn,
                                 const float* __restrict__ en,
                                 float* __restrict__ dist) {
  __shared__ float bsm[2][(KCHUNK / 2) * PRSTRIDE];

  const int lane = threadIdx.x & 31;
  const int wave = threadIdx.x >> 5;
  const int gm0  = blockIdx.y * 128 + wave * 16;
  const int gn0  = blockIdx.x * NBLK;
  const int mrow = gm0 + (lane & 15);
  const int kh   = (lane >> 4) << 1;       // 0 (lanes 0-15) or 2 (lanes 16-31)
  const int prh  = kh >> 1;                // 0 or 1 (K-pair row of my half)
  const int c0   = lane & 15;

  const float* __restrict__ xr = x + (size_t)mrow * DDIM;

  // --- async copy of one 32x128 B chunk into LDS buffer `buf` (interleaved) ---
  // 4096 floats; each of 256 threads issues 16 async b32 transfers with
  // constant-stride address bumps (global +2 rows, LDS +1 pair-row).
  const int tlin = threadIdx.x;
  auto copy_chunk = [&](int k0, int buf) {
    const int k00 = tlin >> 7;             // 0 or 1
    const int col = tlin & 127;
    const float* gsrc = e + (size_t)(k0 + k00) * NE + gn0 + col;
    unsigned ldst = (unsigned)(size_t)&bsm[buf][col * 2 + k00];
    #pragma unroll
    for (int i = 0; i < 16; ++i) {
      asm volatile("global_load_async_to_lds_b32 %0, %1, off"
                   :: "v"(ldst), "v"((unsigned long long)(size_t)gsrc)
                   : "memory");
      gsrc += 2 * NE;                      // +2 K-rows in global
      ldst += PRSTRIDE * 4;                // +1 K-pair row in LDS (bytes)
    }
  };

  v8f acc[8];
  #pragma unroll
  for (int t = 0; t < 8; ++t) acc[t] = (v8f){};

  copy_chunk(0, 0);
  int buf = 0;
  for (int c = 0; c < NCHUNK; ++c) {
    asm volatile("s_wait_asynccnt 0x0" ::: "memory");
    __syncthreads();                       // chunk c visible; prev reads done
    if (c + 1 < NCHUNK) copy_chunk((c + 1) * KCHUNK, buf ^ 1);

    const int k0 = c * KCHUNK;
    const float* __restrict__ bp = &bsm[buf][0];
    const int bbase = c0 * 2;

    // preload all 8 A fragments for this chunk (8 x global b64)
    v2f af[8];
    #pragma unroll
    for (int s = 0; s < 8; ++s) {
      af[s].x = xr[k0 + 4 * s + kh];
      af[s].y = xr[k0 + 4 * s + kh + 1];
    }
    // preload B fragments for first k-step
    v2f bf[8];
    #pragma unroll
    for (int t = 0; t < 8; ++t)
      bf[t] = *(const v2f*)(bp + prh * PRSTRIDE + bbase + t * 32);

    #pragma unroll
    for (int s = 0; s < 8; ++s) {
      v2f bn[8];
      if (s + 1 < 8) {                     // load next step's B while computing
        const int pr = 2 * (s + 1) + prh;
        #pragma unroll
        for (int t = 0; t < 8; ++t)
          bn[t] = *(const v2f*)(bp + pr * PRSTRIDE + bbase + t * 32);
      }
      #pragma unroll
      for (int t = 0; t < 8; ++t)
        acc[t] = __builtin_amdgcn_wmma_f32_16x16x4_f32(
            false, af[s], false, bf[t], (short)0, acc[t], false, false);
      if (s + 1 < 8) {
        #pragma unroll
        for (int t = 0; t < 8; ++t) bf[t] = bn[t];
      }
    }
    buf ^= 1;
  }

  // Epilogue: C/D layout — VGPR r, lanes 0-15: M = gm0+r, lanes 16-31: M = gm0+8+r
  const int mbase = gm0 + ((lane >> 4) << 3);
  for (int t = 0; t < 8; ++t) {
    const int col = gn0 + t * 16 + c0;
    const float ecn = en[col];
    for (int r = 0; r < 8; ++r) {
      const int m = mbase + r;
      dist[(size_t)m * NE + col] = xn[m] + ecn - 2.0f * acc[t][r];
    }
  }
}

// ---------------- top-64 + categorical sample + gather + row MSE ----------------
// One block (256 threads = 8 waves) per row. Each thread caches 32 candidates in
// registers plus its running local min; 64 rounds of (wave32 shuffle arg-min ->
// 8-leader LDS combine). Only the winning thread rescans after consumption.
__global__ void topk_sample_kernel(const float* __restrict__ dist,
                                   const float* __restrict__ e,
                                   const float* __restrict__ x,
                                   float* __restrict__ out_q,
                                   float* __restrict__ out_ind,
                                   float* __restrict__ rowdiff) {
  __shared__ float wv[8];
  __shared__ int   wc[8];
  __shared__ float top_p[TOPK];
  __shared__ int   top_i[TOPK];
  __shared__ int   sel_s;
  __shared__ float red[256];

  const int row  = blockIdx.x;
  const int t    = threadIdx.x;
  const int lane = t & 31;
  const int wid  = t >> 5;
  const float* __restrict__ drow = dist + (size_t)row * NE;
  const float FMAX = 3.402823466e+38f;

  float vals[32];
  #pragma unroll
  for (int j = 0; j < 32; ++j) vals[j] = drow[j * 256 + t];

  unsigned consumed = 0u;
  float bestv = FMAX;
  int   bestj = 0;
  #pragma unroll
  for (int j = 0; j < 32; ++j) {
    if (vals[j] < bestv) { bestv = vals[j]; bestj = j; }
  }

  for (int s = 0; s < TOPK; ++s) {
    // wave32 shuffle arg-min over (value, column)
    float rv_ = bestv;
    int   ci  = bestj * 256 + t;           // global column of my candidate
    #pragma unroll
    for (int off = 16; off > 0; off >>= 1) {
      float ov = __shfl_xor(rv_, off, 32);
      int   oc = __shfl_xor(ci, off, 32);
      if (ov < rv_) { rv_ = ov; ci = oc; }
    }
    if (lane == 0) { wv[wid] = rv_; wc[wid] = ci; }
    __syncthreads();
    // every thread combines the 8 wave minima identically
    float gv = wv[0]; int gc = wc[0];
    #pragma unroll
    for (int w = 1; w < 8; ++w) {
      if (wv[w] < gv) { gv = wv[w]; gc = wc[w]; }
    }
    if (t == (gc & 255)) {                 // winner: consume + rescan local
      consumed |= 1u << (gc >> 8);
      bestv = FMAX; bestj = 0;
      #pragma unroll
      for (int j = 0; j < 32; ++j) {
        bool live = !(consumed & (1u << j));
        if (live && vals[j] < bestv) { bestv = vals[j]; bestj = j; }
      }
    }
    if (t == 0) { top_p[s] = 1.0f / gv; top_i[s] = gc; }
    __syncthreads();                       // protect wv/wc before next round
  }

  if (t == 0) {
    float psum = 0.0f;
    for (int s = 0; s < TOPK; ++s) psum += top_p[s];
    // Deterministic per-row uniform (stand-in for jax categorical, key 42)
    unsigned h = (unsigned)row * 2654435761u + 42u;
    h ^= h >> 16; h *= 0x7feb352du; h ^= h >> 15; h *= 0x846ca68bu; h ^= h >> 16;
    const float u = (float)(h >> 8) * (1.0f / 16777216.0f) * psum;
    float c = 0.0f;
    int sel = top_i[TOPK - 1];
    for (int s = 0; s < TOPK; ++s) {
      c += top_p[s];
      if (u < c) { sel = top_i[s]; break; }
    }
    sel_s = sel;
    out_ind[row] = (float)sel;
  }
  __syncthreads();

  const int sel = sel_s;
  const float q = e[(size_t)t * NE + sel];             // embed.T row gather
  out_q[(size_t)row * DDIM + t] = q;                   // straight-through value
  const float dd = q - x[(size_t)row * DDIM + t];
  red[t] = dd * dd;
  __syncthreads();
  for (int st = 128; st > 0; st >>= 1) {
    if (t < st) red[t] += red[t + st];
    __syncthreads();
  }
  if (t == 0) rowdiff[row] = red[0];
}

// ---------------- deterministic final diff reduction ----------------
__global__ void diff_reduce_kernel(const float* __restrict__ rowdiff,
                                   float* __restrict__ out_diff) {
  __shared__ float red[256];
  float s = 0.0f;
  for (int j = threadIdx.x; j < NROWS; j += 256) s += rowdiff[j];
  red[threadIdx.x] = s;
  __syncthreads();
  for (int st = 128; st > 0; st >>= 1) {
    if ((int)threadIdx.x < st) red[threadIdx.x] += red[threadIdx.x + st];
    __syncthreads();
  }
  if (threadIdx.x == 0) out_diff[0] = red[0] * (1.0f / 8388608.0f);
}

extern "C" void kernel_launch(void* const* d_in, const int* in_sizes, int n_in,
                              void* d_out, int out_size, void* d_ws, size_t ws_size,
                              hipStream_t stream) {
  (void)in_sizes; (void)n_in; (void)out_size; (void)ws_size;
  const float* x = (const float*)d_in[0];   // [8,64,64,256] -> [32768,256]
  const float* e = (const float*)d_in[1];   // [256,8192]
  float* out = (float*)d_out;

  float* xn      = (float*)d_ws;            // 32768 floats
  float* en      = xn + NROWS;              // 8192 floats
  float* rowdiff = en + NE;                 // 32768 floats

  float* q    = out + Q_OFF;
  float* diff = out + DIFF_OFF;
  float* ind  = out + IND_OFF;
  float* dist = out + DIST_OFF;

  row_norm_kernel<<<NROWS, 256, 0, stream>>>(x, xn);
  col_norm_kernel<<<NE / 256, 256, 0, stream>>>(e, en);

  dim3 g(NE / NBLK, NROWS / 128);
  dist_wmma_kernel<<<g, 256, 0, stream>>>(x, e, xn, en, dist);

  topk_sample_kernel<<<NROWS, 256, 0, stream>>>(dist, e, x, q, ind, rowdiff);
  diff_reduce_kernel<<<1, 256, 0, stream>>>(rowdiff, diff);
}